// cvx_43593918054943
// MI455X (gfx1250) — compile-verified
//
#include <hip/hip_runtime.h>
#include <hip/hip_bf16.h>
#include <cstdint>
#include <cstddef>

// ---------------------------------------------------------------------------
// GCN pipeline for MI455X (gfx1250, wave32).
// Dense transforms use V_WMMA_F32_16X16X4_F32 (fp32-exact matrix core path).
// Edge aggregation uses float4 gathers + grouped f32 atomics (bandwidth bound).
// ---------------------------------------------------------------------------

typedef float v2f __attribute__((ext_vector_type(2)));
typedef float v8f __attribute__((ext_vector_type(8)));

#if defined(__has_builtin)
#  if __has_builtin(__builtin_amdgcn_wmma_f32_16x16x4_f32)
#    define HAVE_WMMA_F32X4 1
#  endif
#endif
#ifndef HAVE_WMMA_F32X4
#  define HAVE_WMMA_F32X4 0
#endif

// ---------------------------------------------------------------------------
// Degree kernels
// ---------------------------------------------------------------------------
__global__ void k_deg_init(float* __restrict__ deg, int n) {
    int i = blockIdx.x * blockDim.x + threadIdx.x;
    if (i < n) deg[i] = 1.0f;                      // self-loop contributes 1
}

__global__ void k_deg_accum(const int* __restrict__ dst, float* __restrict__ deg, int ne) {
    int e = blockIdx.x * blockDim.x + threadIdx.x;
    if (e < ne) atomicAdd(&deg[dst[e]], 1.0f);
}

__global__ void k_deg_fin(const float* __restrict__ deg, float* __restrict__ dinvs,
                          float* __restrict__ dinv, int n) {
    int i = blockIdx.x * blockDim.x + threadIdx.x;
    if (i < n) {
        float d = deg[i];
        dinvs[i] = rsqrtf(d);
        dinv[i]  = 1.0f / d;
    }
}

// ---------------------------------------------------------------------------
// WMMA GEMM:  out[nrows, NOUT] = act(A[nrows, K] @ W[K, NOUT] (+ bias))
// 256 threads = 8 waves. Each wave computes one 16x16 tile.
//   CT = col tiles (NOUT/16 rounded up), RT = 8/CT row tiles per block.
// A-operand layout (16x4 fp32): lanes 0-15 carry K=k..k+1, lanes 16-31 K=k+2..k+3.
// B-operand symmetric (N across lanes 0-15). D: VGPR r -> (m = r + 8*hi, n = lane&15).
// ---------------------------------------------------------------------------
template <int K, int NOUT, bool BIAS, bool RELU>
__global__ void k_gemm(const float* __restrict__ A, const float* __restrict__ W,
                       const float* __restrict__ bias, float* __restrict__ out, int nrows) {
    constexpr int CT = (NOUT + 15) / 16;
    constexpr int RT = 8 / CT;
    const int wave = threadIdx.x >> 5;
    const int lane = threadIdx.x & 31;
    const int rt = wave / CT;
    const int ct = wave % CT;
    const int row0 = blockIdx.x * (RT * 16) + rt * 16;
    const int col0 = ct * 16;
    const int hi  = lane >> 4;      // 0: lanes 0-15, 1: lanes 16-31
    const int l15 = lane & 15;

#if HAVE_WMMA_F32X4
    // Clamp the A row so every lane issues valid loads (EXEC must be all-1s at WMMA).
    int arow  = row0 + l15;
    int arowc = (arow < nrows) ? arow : (nrows - 1);
    const float* aptr = A + (size_t)arowc * K + 2 * hi;
    const int n = col0 + l15;

    v8f c = {0.f, 0.f, 0.f, 0.f, 0.f, 0.f, 0.f, 0.f};
#pragma unroll
    for (int k = 0; k < K; k += 4) {
        v2f a;
        a.x = aptr[k + 0];
        a.y = aptr[k + 1];
        v2f b;
        const int kb = k + 2 * hi;
        if constexpr (NOUT % 16 == 0) {
            b.x = W[(size_t)(kb + 0) * NOUT + n];
            b.y = W[(size_t)(kb + 1) * NOUT + n];
        } else {
            b.x = (n < NOUT) ? W[(size_t)(kb + 0) * NOUT + n] : 0.0f;
            b.y = (n < NOUT) ? W[(size_t)(kb + 1) * NOUT + n] : 0.0f;
        }
        c = __builtin_amdgcn_wmma_f32_16x16x4_f32(false, a, false, b, (short)0, c,
                                                  false, false);
    }
#pragma unroll
    for (int r = 0; r < 8; ++r) {
        int m = row0 + r + 8 * hi;
        if (m < nrows && (NOUT % 16 == 0 || n < NOUT)) {
            float v = c[r];
            if (BIAS) v += bias[n];
            if (RELU) v = fmaxf(v, 0.0f);
            out[(size_t)m * NOUT + n] = v;
        }
    }
#else
    // Scalar fallback (only if the fp32 WMMA builtin is unavailable).
    (void)wave; (void)lane; (void)rt; (void)ct; (void)col0; (void)hi; (void)l15;
    const int base = blockIdx.x * (RT * 16);
    for (int idx = threadIdx.x; idx < RT * 16 * NOUT; idx += blockDim.x) {
        int m = base + idx / NOUT;
        int nn = idx % NOUT;
        if (m < nrows) {
            float acc = 0.0f;
            for (int k = 0; k < K; ++k) acc = fmaf(A[(size_t)m * K + k], W[(size_t)k * NOUT + nn], acc);
            if (BIAS) acc += bias[nn];
            if (RELU) acc = fmaxf(acc, 0.0f);
            out[(size_t)m * NOUT + nn] = acc;
        }
    }
#endif
}

template <int K, int NOUT, bool BIAS, bool RELU>
static void launch_gemm(const float* A, const float* W, const float* bias, float* out,
                        int nrows, hipStream_t stream) {
    constexpr int CT = (NOUT + 15) / 16;
    constexpr int ROWS = (8 / CT) * 16;
    int grid = (nrows + ROWS - 1) / ROWS;
    k_gemm<K, NOUT, BIAS, RELU><<<grid, 256, 0, stream>>>(A, W, bias, out, nrows);
}

// ---------------------------------------------------------------------------
// Edge scatter:  agg[dst] += hw[src] * (dinvs[src]*dinvs[dst])
// One thread per (edge, 4-channel group): float4 gather, 4 f32 atomics.
// ---------------------------------------------------------------------------
template <int NOUT>
__global__ void k_scatter(const float* __restrict__ hw, const int* __restrict__ src,
                          const int* __restrict__ dst, const float* __restrict__ dinvs,
                          float* __restrict__ agg, int nedges) {
    constexpr int G = NOUT / 4;
    long long idx = (long long)blockIdx.x * blockDim.x + threadIdx.x;
    if (idx >= (long long)nedges * G) return;
    int e = (int)(idx / G);
    int g = (int)(idx % G);
    int s = src[e];
    int d = dst[e];
    float norm = dinvs[s] * dinvs[d];
    float4 v = *(const float4*)(hw + (size_t)s * NOUT + g * 4);
    float* a = agg + (size_t)d * NOUT + g * 4;
    atomicAdd(a + 0, v.x * norm);
    atomicAdd(a + 1, v.y * norm);
    atomicAdd(a + 2, v.z * norm);
    atomicAdd(a + 3, v.w * norm);
}

// out = act(agg + hw * deg_inv + bias)   (self-loop term + bias + optional relu)
template <int NOUT, bool RELU>
__global__ void k_finalize(const float* __restrict__ hw, const float* __restrict__ agg,
                           const float* __restrict__ dinv, const float* __restrict__ bias,
                           float* __restrict__ out, int n) {
    long long i = (long long)blockIdx.x * blockDim.x + threadIdx.x;
    if (i >= (long long)n * NOUT) return;
    int node = (int)(i / NOUT);
    int c    = (int)(i % NOUT);
    float v = agg[i] + hw[i] * dinv[node] + bias[c];
    if (RELU) v = fmaxf(v, 0.0f);
    out[i] = v;
}

// ---------------------------------------------------------------------------
// Heads
// ---------------------------------------------------------------------------
__global__ void k_edge_head(const float* __restrict__ h3, const int* __restrict__ src,
                            const int* __restrict__ dst, const float* __restrict__ Ws1,
                            const float* __restrict__ bs1, const float* __restrict__ Ws2,
                            const float* __restrict__ bs2, float* __restrict__ out,
                            int nedges) {
    int e = blockIdx.x * blockDim.x + threadIdx.x;
    if (e >= nedges) return;
    int s = src[e], d = dst[e];
    const float4* ps = (const float4*)(h3 + (size_t)s * 8);
    const float4* pd = (const float4*)(h3 + (size_t)d * 8);
    float4 a0 = ps[0], a1 = ps[1], b0 = pd[0], b1 = pd[1];
    float emb[16] = {a0.x, a0.y, a0.z, a0.w, a1.x, a1.y, a1.z, a1.w,
                     b0.x, b0.y, b0.z, b0.w, b1.x, b1.y, b1.z, b1.w};
    float acc = bs2[0];
#pragma unroll
    for (int j = 0; j < 8; ++j) {
        float t = bs1[j];
#pragma unroll
        for (int k = 0; k < 16; ++k) t = fmaf(emb[k], Ws1[k * 8 + j], t);
        t = fmaxf(t, 0.0f);
        acc = fmaf(t, Ws2[j], acc);
    }
    out[e] = 1.0f / (1.0f + expf(-acc));
}

__global__ void k_node_head(const float* __restrict__ h3, const float* __restrict__ Wv1,
                            const float* __restrict__ bv1, const float* __restrict__ Wv2,
                            const float* __restrict__ bv2, float* __restrict__ out, int n) {
    int i = blockIdx.x * blockDim.x + threadIdx.x;
    if (i >= n) return;
    const float4* p = (const float4*)(h3 + (size_t)i * 8);
    float4 h0 = p[0], h1 = p[1];
    float hv[8] = {h0.x, h0.y, h0.z, h0.w, h1.x, h1.y, h1.z, h1.w};
    float acc = bv2[0];
#pragma unroll
    for (int j = 0; j < 4; ++j) {
        float t = bv1[j];
#pragma unroll
        for (int k = 0; k < 8; ++k) t = fmaf(hv[k], Wv1[k * 4 + j], t);
        t = fmaxf(t, 0.0f);
        acc = fmaf(t, Wv2[j], acc);
    }
    float sg = 1.0f / (1.0f + expf(-acc));
    float vp = 0.9f + 0.2f * sg;
    out[i] = vp * vp;
}

// ---------------------------------------------------------------------------
// Launcher
// ---------------------------------------------------------------------------
extern "C" void kernel_launch(void* const* d_in, const int* in_sizes, int n_in,
                              void* d_out, int out_size, void* d_ws, size_t ws_size,
                              hipStream_t stream) {
    (void)n_in; (void)out_size; (void)ws_size;

    const float* x     = (const float*)d_in[0];
    const int*   ei    = (const int*)d_in[1];
    const float* W_enc = (const float*)d_in[2];
    const float* b_enc = (const float*)d_in[3];
    const float* W_g1  = (const float*)d_in[4];
    const float* b_g1  = (const float*)d_in[5];
    const float* W_g2  = (const float*)d_in[6];
    const float* b_g2  = (const float*)d_in[7];
    const float* W_g3  = (const float*)d_in[8];
    const float* b_g3  = (const float*)d_in[9];
    const float* W_s1  = (const float*)d_in[10];
    const float* b_s1  = (const float*)d_in[11];
    const float* W_s2  = (const float*)d_in[12];
    const float* b_s2  = (const float*)d_in[13];
    const float* W_v1  = (const float*)d_in[14];
    const float* b_v1  = (const float*)d_in[15];
    const float* W_v2  = (const float*)d_in[16];
    const float* b_v2  = (const float*)d_in[17];

    const int N = in_sizes[0] / 32;
    const int E = in_sizes[1] / 2;
    const int* src = ei;
    const int* dst = ei + E;

    // Workspace layout (floats): deg | dinvs | dinv | H (N*64, reused) | HW (N*32) | AGG (N*32)
    float* ws    = (float*)d_ws;
    float* deg   = ws;
    float* dinvs = deg + N;
    float* dinv  = dinvs + N;
    float* H     = dinv + N;
    float* HW    = H + (size_t)N * 64;
    float* AGG   = HW + (size_t)N * 32;

    const int B = 256;

    // Degrees (with self-loop).
    k_deg_init<<<(N + B - 1) / B, B, 0, stream>>>(deg, N);
    k_deg_accum<<<(E + B - 1) / B, B, 0, stream>>>(dst, deg, E);
    k_deg_fin<<<(N + B - 1) / B, B, 0, stream>>>(deg, dinvs, dinv, N);

    // Encoder: h = relu(relu(x @ W_enc + b_enc)) == relu(x @ W_enc + b_enc)
    launch_gemm<32, 64, true, true>(x, W_enc, b_enc, H, N, stream);

    // GCN layer 1: 64 -> 32
    launch_gemm<64, 32, false, false>(H, W_g1, nullptr, HW, N, stream);
    hipMemsetAsync(AGG, 0, (size_t)N * 32 * sizeof(float), stream);
    {
        long long tot = (long long)E * (32 / 4);
        k_scatter<32><<<(unsigned)((tot + B - 1) / B), B, 0, stream>>>(HW, src, dst, dinvs, AGG, E);
        long long tn = (long long)N * 32;
        k_finalize<32, true><<<(unsigned)((tn + B - 1) / B), B, 0, stream>>>(HW, AGG, dinv, b_g1, H, N);
    }

    // GCN layer 2: 32 -> 16
    launch_gemm<32, 16, false, false>(H, W_g2, nullptr, HW, N, stream);
    hipMemsetAsync(AGG, 0, (size_t)N * 16 * sizeof(float), stream);
    {
        long long tot = (long long)E * (16 / 4);
        k_scatter<16><<<(unsigned)((tot + B - 1) / B), B, 0, stream>>>(HW, src, dst, dinvs, AGG, E);
        long long tn = (long long)N * 16;
        k_finalize<16, true><<<(unsigned)((tn + B - 1) / B), B, 0, stream>>>(HW, AGG, dinv, b_g2, H, N);
    }

    // GCN layer 3: 16 -> 8 (no relu)
    launch_gemm<16, 8, false, false>(H, W_g3, nullptr, HW, N, stream);
    hipMemsetAsync(AGG, 0, (size_t)N * 8 * sizeof(float), stream);
    {
        long long tot = (long long)E * (8 / 4);
        k_scatter<8><<<(unsigned)((tot + B - 1) / B), B, 0, stream>>>(HW, src, dst, dinvs, AGG, E);
        long long tn = (long long)N * 8;
        k_finalize<8, false><<<(unsigned)((tn + B - 1) / B), B, 0, stream>>>(HW, AGG, dinv, b_g3, H, N);
    }

    // Heads: d_out = [switch_scores (E) | v_sq_pred (N)]
    float* out_sw = (float*)d_out;
    float* out_v  = out_sw + E;
    k_edge_head<<<(E + B - 1) / B, B, 0, stream>>>(H, src, dst, W_s1, b_s1, W_s2, b_s2, out_sw, E);
    k_node_head<<<(N + B - 1) / B, B, 0, stream>>>(H, W_v1, b_v1, W_v2, b_v2, out_v, N);
}